// RNNModel_65901978190205
// MI455X (gfx1250) — compile-verified
//
#include <hip/hip_runtime.h>

#define HID 18
#define KD 24   // padded K for LSTM GEMM: 5 (x) + 18 (h) + 1 pad
#define ND 80   // padded N: 72 -> 80

typedef __attribute__((ext_vector_type(2))) float v2f;
typedef __attribute__((ext_vector_type(8))) float v8f;

__device__ __forceinline__ float wave_sum(float x) {
#pragma unroll
  for (int off = 16; off > 0; off >>= 1) x += __shfl_xor(x, off, 32);
  return x;
}

// v = dm @ conj(u); dm = 2x2 complex row-major interleaved {r00,i00,r01,i01,r10,i10,r11,i11}
__device__ __forceinline__ void dm_conj_u(const float d[8], const float pr[2],
                                          const float pi[2], float v[4]) {
  v[0] = d[0] * pr[0] + d[1] * pi[0] + d[2] * pr[1] + d[3] * pi[1];
  v[1] = d[1] * pr[0] - d[0] * pi[0] + d[3] * pr[1] - d[2] * pi[1];
  v[2] = d[4] * pr[0] + d[5] * pi[0] + d[6] * pr[1] + d[7] * pi[1];
  v[3] = d[5] * pr[0] - d[4] * pi[0] + d[7] * pr[1] - d[6] * pi[1];
}
__device__ __forceinline__ float dot_re(const float pr[2], const float pi[2], const float v[4]) {
  return pr[0] * v[0] - pi[0] * v[1] + pr[1] * v[2] - pi[1] * v[3];
}
__device__ __forceinline__ float dot_im(const float pr[2], const float pi[2], const float v[4]) {
  return pr[0] * v[1] + pi[0] * v[0] + pr[1] * v[3] + pi[1] * v[2];
}

// Evolve both pure-state vectors through the 18-parameter circuit.
// lane i holds amplitude i; wire w <-> lane bit (16 >> w).
__device__ __forceinline__ void evolve(const float th[18], int lane, float pr[2], float pi[2]) {
  const int bits[5] = {16, 8, 4, 2, 1};
  // RX layer: [[c, -i s], [-i s, c]]
#pragma unroll
  for (int w = 0; w < 5; ++w) {
    float c = cosf(0.5f * th[w]), s = sinf(0.5f * th[w]);
    int bit = bits[w];
#pragma unroll
    for (int a = 0; a < 2; ++a) {
      float br = __shfl_xor(pr[a], bit, 32);
      float bi = __shfl_xor(pi[a], bit, 32);
      float nr = c * pr[a] + s * bi;
      float ni = c * pi[a] - s * br;
      pr[a] = nr; pi[a] = ni;
    }
  }
  // RY layer: [[c, -s], [s, c]]
#pragma unroll
  for (int w = 0; w < 5; ++w) {
    float c = cosf(0.5f * th[5 + w]), s = sinf(0.5f * th[5 + w]);
    int bit = bits[w];
    float sg = (lane & bit) ? s : -s;
#pragma unroll
    for (int a = 0; a < 2; ++a) {
      float br = __shfl_xor(pr[a], bit, 32);
      float bi = __shfl_xor(pi[a], bit, 32);
      float nr = c * pr[a] + sg * br;
      float ni = c * pi[a] + sg * bi;
      pr[a] = nr; pi[a] = ni;
    }
  }
  // CRX layer: RX on target iff control bit set
  const int cw[8] = {0, 0, 0, 0, 1, 2, 3, 4};
  const int tw[8] = {1, 2, 3, 4, 0, 0, 0, 0};
#pragma unroll
  for (int k = 0; k < 8; ++k) {
    float c = cosf(0.5f * th[10 + k]), s = sinf(0.5f * th[10 + k]);
    int cbit = bits[cw[k]], tbit = bits[tw[k]];
    bool act = (lane & cbit) != 0;
#pragma unroll
    for (int a = 0; a < 2; ++a) {
      float br = __shfl_xor(pr[a], tbit, 32);
      float bi = __shfl_xor(pi[a], tbit, 32);
      float nr = c * pr[a] + s * bi;
      float ni = c * pi[a] - s * br;
      pr[a] = act ? nr : pr[a];
      pi[a] = act ? ni : pi[a];
    }
  }
}

// Projective measurement on mask ((lane & maskbit)==0); folds 1/p into dm. Returns +-1.
__device__ __forceinline__ float do_measure(int maskbit, float u, int lane,
                                            float pr[2], float pi[2], float d[8]) {
  float v[4];
  dm_conj_u(d, pr, pi, v);
  float diag = dot_re(pr, pi, v);  // Re rho[i,i]
  float m = ((lane & maskbit) == 0) ? 1.0f : 0.0f;
  float p = wave_sum(m * diag);
  p = fminf(fmaxf(p, 1e-7f), 1.0f - 1e-7f);
  bool down = (u >= p);
  float keep = down ? (1.0f - m) : m;
  pr[0] *= keep; pi[0] *= keep; pr[1] *= keep; pi[1] *= keep;
  float inv = 1.0f / (down ? (1.0f - p) : p);
#pragma unroll
  for (int q = 0; q < 8; ++q) d[q] *= inv;
  return down ? -1.0f : 1.0f;
}

// ---------------- kernels ----------------

__global__ void k_init(const float* __restrict__ ini_theta, const float* __restrict__ ip,
                       const float* __restrict__ iup, float* __restrict__ dm,
                       float* __restrict__ iu, float* __restrict__ xh,
                       float* __restrict__ cst, int B) {
  int b = blockIdx.x * blockDim.x + threadIdx.x;
  if (b >= B) return;
  const float PI = 3.14159265358979323846f;
  float s = 0.0f;
  for (int k = 0; k < 5; ++k) s += ip[b * 5 + k];
  float d[8] = {0, 0, 0, 0, 0, 0, 0, 0};
  for (int k = 0; k < 5; ++k) {
    float h0 = ini_theta[(b * 5 + k) * 2 + 0] * PI;
    float h1 = ini_theta[(b * 5 + k) * 2 + 1] * PI;
    float c0 = cosf(h0), s0 = sinf(h0), c1 = cosf(h1), s1 = sinf(h1);
    // psi = [e^{-i h1} cos h0, -i e^{+i h1} sin h0]
    float ar = c1 * c0, ai = -s1 * c0;
    float br = s1 * s0, bi = -c1 * s0;
    float w = ip[b * 5 + k] / s;
    d[0] += w * (ar * ar + ai * ai);
    d[2] += w * (ar * br + ai * bi);
    d[3] += w * (ai * br - ar * bi);
    d[4] += w * (br * ar + bi * ai);
    d[5] += w * (bi * ar - br * ai);
    d[6] += w * (br * br + bi * bi);
  }
  for (int q = 0; q < 8; ++q) dm[b * 8 + q] = d[q];
  float u0 = iup[b * 3 + 0] * 2.0f - 1.0f;
  float u1 = iup[b * 3 + 1] * 2.0f - 1.0f;
  float u2 = iup[b * 3 + 2] * 2.0f - 1.0f;
  iu[b * 3 + 0] = u0; iu[b * 3 + 1] = u1; iu[b * 3 + 2] = u2;
  for (int j = 0; j < KD; ++j) xh[b * KD + j] = 0.0f;
  xh[b * KD + 0] = u0; xh[b * KD + 1] = u1; xh[b * KD + 2] = u2;
  for (int j = 0; j < HID; ++j) cst[b * HID + j] = 0.0f;
}

// Pack per-step weights into padded (KD x ND): rows 0..indim-1 = Wx, 5..22 = Wh, rest 0.
__global__ void k_pack(const float* Wx1, const float* Wh1, const float* Wx2, const float* Wh2,
                       const float* Wx3, const float* Wh3, const float* Wx4, const float* Wh4,
                       const float* Wx5, const float* Wh5, float* __restrict__ wp) {
  int t = blockIdx.x * blockDim.x + threadIdx.x;
  if (t >= 5 * KD * ND) return;
  int m = t / (KD * ND);
  int rem = t % (KD * ND);
  int r = rem / ND, n = rem % ND;
  const float* Wx[5] = {Wx1, Wx2, Wx3, Wx4, Wx5};
  const float* Wh[5] = {Wh1, Wh2, Wh3, Wh4, Wh5};
  const int indim[5] = {3, 5, 5, 5, 5};
  float v = 0.0f;
  if (n < 72) {
    if (r < indim[m]) v = Wx[m][r * 72 + n];
    else if (r >= 5 && r < 23) v = Wh[m][(r - 5) * 72 + n];
  }
  wp[t] = v;
}

// z[B x ND] = xh[B x KD] @ wp[KD x ND] via V_WMMA_F32_16X16X4_F32.
// One 16x16 tile per wave; A layout: lane(m,half) -> {A[m][k0], A[m][k0+1]}, k0 = 4kb+2*half.
__global__ void k_lstm_gemm(const float* __restrict__ xh, const float* __restrict__ wp,
                            float* __restrict__ z, int tilesM) {
  int wave = (blockIdx.x * blockDim.x + threadIdx.x) >> 5;
  int lane = threadIdx.x & 31;
  if (wave >= tilesM * 5) return;
  int tM = wave / 5, tN = wave % 5;
  int m = lane & 15, half = lane >> 4;
  const float* arow = xh + (size_t)(tM * 16 + m) * KD;
  const float* bcol = wp + tN * 16 + m;
  v8f acc = {0.0f, 0.0f, 0.0f, 0.0f, 0.0f, 0.0f, 0.0f, 0.0f};
#pragma unroll
  for (int kb = 0; kb < KD / 4; ++kb) {
    int k0 = kb * 4 + half * 2;
    v2f a;  a.x  = arow[k0];                 a.y  = arow[k0 + 1];
    v2f bm; bm.x = bcol[(size_t)k0 * ND];    bm.y = bcol[(size_t)(k0 + 1) * ND];
    acc = __builtin_amdgcn_wmma_f32_16x16x4_f32(false, a, false, bm, (short)0, acc,
                                                false, false);
  }
#pragma unroll
  for (int r = 0; r < 8; ++r) {
    int row = tM * 16 + r + half * 8;  // C/D layout: VGPR r = rows r / r+8
    z[(size_t)row * ND + tN * 16 + m] = acc[r];
  }
}

__global__ void k_lstm_act(const float* __restrict__ z, const float* __restrict__ bias,
                           float* __restrict__ cst, float* __restrict__ hbuf,
                           float* __restrict__ xh, int B) {
  int t = blockIdx.x * blockDim.x + threadIdx.x;
  if (t >= B * HID) return;
  int b = t / HID, j = t % HID;
  const float* zr = z + (size_t)b * ND;
  float iv = zr[j]      + bias[j];
  float fv = zr[18 + j] + bias[18 + j];
  float gv = zr[36 + j] + bias[36 + j];
  float ov = zr[54 + j] + bias[54 + j];
  float si = 1.0f / (1.0f + expf(-iv));
  float sf = 1.0f / (1.0f + expf(-fv));
  float so = 1.0f / (1.0f + expf(-ov));
  float c2 = sf * cst[t] + si * tanhf(gv);
  float h2 = so * tanhf(c2);
  cst[t] = c2;
  hbuf[t] = h2;
  xh[(size_t)b * KD + 5 + j] = h2;  // h feeds next GEMM
}

// One wave per batch element: circuit + two measurements + partial trace -> new dm, next x.
__global__ void k_circuit(const float* __restrict__ hbuf, const float* __restrict__ um,
                          const float* __restrict__ iu, float* __restrict__ dm,
                          float* __restrict__ xh, int step, int B) {
  int wave = (blockIdx.x * blockDim.x + threadIdx.x) >> 5;
  int lane = threadIdx.x & 31;
  if (wave >= B) return;
  int b = wave;
  float th[18];
#pragma unroll
  for (int k = 0; k < 18; ++k) th[k] = hbuf[b * 18 + k];
  float d[8];
#pragma unroll
  for (int q = 0; q < 8; ++q) d[q] = dm[b * 8 + q];
  float pr[2], pi[2];
  pr[0] = (lane == 0)  ? 1.0f : 0.0f; pi[0] = 0.0f;  // |00000>
  pr[1] = (lane == 16) ? 1.0f : 0.0f; pi[1] = 0.0f;  // |10000>
  evolve(th, lane, pr, pi);
  float v1 = do_measure(16, um[b * 8 + step * 2 + 0], lane, pr, pi, d);  // wire 0
  float v2 = do_measure(8,  um[b * 8 + step * 2 + 1], lane, pr, pi, d);  // wire 1
  // partial trace over wires 1..4: dm'[r][s] = sum_k rho[16r+k, 16s+k]
  float v[4];
  dm_conj_u(d, pr, pi, v);
  float vo[4];
#pragma unroll
  for (int q = 0; q < 4; ++q) vo[q] = __shfl_xor(v[q], 16, 32);
  float dsr = dot_re(pr, pi, v),  dsi = dot_im(pr, pi, v);
  float dcr = dot_re(pr, pi, vo), dci = dot_im(pr, pi, vo);
  int hi = (lane >> 4) & 1;
  float nd[8];
  nd[0] = wave_sum(hi ? 0.0f : dsr);
  nd[1] = wave_sum(hi ? 0.0f : dsi);
  nd[2] = wave_sum(hi ? 0.0f : dcr);
  nd[3] = wave_sum(hi ? 0.0f : dci);
  nd[4] = wave_sum(hi ? dcr : 0.0f);
  nd[5] = wave_sum(hi ? dci : 0.0f);
  nd[6] = wave_sum(hi ? dsr : 0.0f);
  nd[7] = wave_sum(hi ? dsi : 0.0f);
  if (lane == 0) {
#pragma unroll
    for (int q = 0; q < 8; ++q) dm[b * 8 + q] = nd[q];
    xh[(size_t)b * KD + 0] = v1;
    xh[(size_t)b * KD + 1] = v2;
    xh[(size_t)b * KD + 2] = iu[b * 3 + 0];
    xh[(size_t)b * KD + 3] = iu[b * 3 + 1];
    xh[(size_t)b * KD + 4] = iu[b * 3 + 2];
  }
}

// Final circuit + Pauli-sum expectations + loss.
__global__ void k_final(const float* __restrict__ hbuf, const float* __restrict__ dm,
                        const float* __restrict__ iu, float* __restrict__ out, int B) {
  int wave = (blockIdx.x * blockDim.x + threadIdx.x) >> 5;
  int lane = threadIdx.x & 31;
  if (wave >= B) return;
  int b = wave;
  float th[18];
#pragma unroll
  for (int k = 0; k < 18; ++k) th[k] = hbuf[b * 18 + k];
  float d[8];
#pragma unroll
  for (int q = 0; q < 8; ++q) d[q] = dm[b * 8 + q];
  float pr[2], pi[2];
  pr[0] = (lane == 0)  ? 1.0f : 0.0f; pi[0] = 0.0f;
  pr[1] = (lane == 16) ? 1.0f : 0.0f; pi[1] = 0.0f;
  evolve(th, lane, pr, pi);
  float v[4];
  dm_conj_u(d, pr, pi, v);
  // tz: sum_i Re(rho[i,i]) * sum_w (-1)^{bit_w(i)}
  float zc = 5.0f - 2.0f * (float)__popc(lane & 31);
  float tz = wave_sum(zc * dot_re(pr, pi, v));
  float txl = 0.0f, tyl = 0.0f;
  const int bits[5] = {16, 8, 4, 2, 1};
#pragma unroll
  for (int w = 0; w < 5; ++w) {
    int bit = bits[w];
    float vs[4];
#pragma unroll
    for (int q = 0; q < 4; ++q) vs[q] = __shfl_xor(v[q], bit, 32);
    float zr = dot_re(pr, pi, vs);   // rho[i, i^bit]
    float zi = dot_im(pr, pi, vs);
    txl += zr;                                 // X: entry 1
    tyl += (lane & bit) ? zi : -zi;            // Y: entry +-i
  }
  float tx = wave_sum(txl), ty = wave_sum(tyl);
  if (lane == 0) {
    float u0 = iu[b * 3 + 0], u1 = iu[b * 3 + 1], u2 = iu[b * 3 + 2];
    out[b] = u0 * tx + u1 * ty + u2 * tz + sqrtf(u0 * u0 + u1 * u1 + u2 * u2) - 1.0f;
  }
}

extern "C" void kernel_launch(void* const* d_in, const int* in_sizes, int n_in,
                              void* d_out, int out_size, void* d_ws, size_t ws_size,
                              hipStream_t stream) {
  (void)n_in; (void)out_size; (void)ws_size;
  const float* ini_theta = (const float*)d_in[0];
  const float* ip  = (const float*)d_in[1];
  const float* iup = (const float*)d_in[2];
  const float* um  = (const float*)d_in[3];
  const float* Wx[5]; const float* Wh[5]; const float* bs[5];
  for (int m = 0; m < 5; ++m) {
    Wx[m] = (const float*)d_in[4 + 3 * m];
    Wh[m] = (const float*)d_in[5 + 3 * m];
    bs[m] = (const float*)d_in[6 + 3 * m];
  }
  int B = in_sizes[0] / 10;  // ini_theta is (B,5,2)

  float* ws = (float*)d_ws;
  float* dm = ws;                        // B*8  : 2x2 complex density matrix (interleaved)
  float* iu = dm + (size_t)B * 8;        // B*3
  float* xh = iu + (size_t)B * 3;        // B*KD : GEMM A matrix [x | h | pad]
  float* zb = xh + (size_t)B * KD;       // B*ND : GEMM output
  float* hb = zb + (size_t)B * ND;       // B*HID
  float* cb = hb + (size_t)B * HID;      // B*HID
  float* wp = cb + (size_t)B * HID;      // 5*KD*ND packed weights

  k_init<<<(B + 255) / 256, 256, 0, stream>>>(ini_theta, ip, iup, dm, iu, xh, cb, B);
  k_pack<<<(5 * KD * ND + 255) / 256, 256, 0, stream>>>(
      Wx[0], Wh[0], Wx[1], Wh[1], Wx[2], Wh[2], Wx[3], Wh[3], Wx[4], Wh[4], wp);

  int tilesM = B / 16;
  int gemmThreads = tilesM * 5 * 32;
  for (int m = 0; m < 5; ++m) {
    k_lstm_gemm<<<(gemmThreads + 255) / 256, 256, 0, stream>>>(xh, wp + m * KD * ND, zb, tilesM);
    k_lstm_act<<<(B * HID + 255) / 256, 256, 0, stream>>>(zb, bs[m], cb, hb, xh, B);
    if (m < 4)
      k_circuit<<<(B * 32 + 255) / 256, 256, 0, stream>>>(hb, um, iu, dm, xh, m, B);
  }
  k_final<<<(B * 32 + 255) / 256, 256, 0, stream>>>(hb, dm, iu, (float*)d_out, B);
}